// PlanarScannerNet_12902081757868
// MI455X (gfx1250) — compile-verified
//
#include <hip/hip_runtime.h>
#include <math.h>

typedef float v2f __attribute__((ext_vector_type(2)));
typedef float v8f __attribute__((ext_vector_type(8)));

#define ROWS    64
#define MLEN    128
#define HBR     512
#define HLAT    2048
#define OUTW    1536   // LEAF*LATENT
#define LATENT  48

// ---------------------------------------------------------------------------
// Kernel 1: branch net + gates + router (one int per row -> d_ws)
// h = relu(main @ Wb1 + bb1) built by WMMA into LDS; z = h @ Wb2 + bb2 by a
// float4 per-row dot; p = softmax(z); gates/argmax (ties -> largest index).
// ---------------------------------------------------------------------------
__global__ __launch_bounds__(256) void router_kernel(
    const float* __restrict__ batch,   // (B, 2, 128)
    const float* __restrict__ Wb1,     // (128, 512)
    const float* __restrict__ bb1,     // (512)
    const float* __restrict__ Wb2,     // (512, 2)
    const float* __restrict__ bb2,     // (2)
    int* __restrict__ router)          // (B)
{
    __shared__ v2f   XsTp[MLEN / 2][80]; // main cam, K-paired transpose:
                                         // XsTp[k/2][m] = {x[k][m], x[k+1][m]}
    __shared__ float Hs[ROWS][516];      // h tile; 16B rows, bank-clean writes
    __shared__ float Wb2s[HBR * 2];      // Wb2 copy (4 KB)
    __shared__ float Zq[4][ROWS][2];     // per-quarter z partials

    const int tid  = threadIdx.x;
    const int lane = tid & 31;
    const int w    = tid >> 5;
    const int r0   = blockIdx.x * ROWS;

    // stage main-cam rows (coalesced float4 loads) into K-paired transpose
    #pragma unroll
    for (int i = 0; i < 8; ++i) {
        const int f  = tid + i * 256;          // 2048 float4s total
        const int m  = f >> 5;                 // 32 float4 per row
        const int c4 = f & 31;
        const float4 v = *(const float4*)(batch + (size_t)(r0 + m) * 256 + c4 * 4);
        XsTp[c4 * 2 + 0][m] = (v2f){v.x, v.y};
        XsTp[c4 * 2 + 1][m] = (v2f){v.z, v.w};
    }
    for (int i = tid; i < HBR * 2; i += 256) Wb2s[i] = Wb2[i];
    __syncthreads();

    const int half = lane >> 4;   // K sub-offset selector for wmma fragments
    const int l16  = lane & 15;

    // wave w handles col-tiles {w, w+8, w+16, w+24}; all 4 row tiles each
    for (int j = 0; j < 4; ++j) {
        const int n = (w + 8 * j) * 16 + l16;  // this lane's output column
        const float bias = bb1[n];

        v8f acc[4];
        #pragma unroll
        for (int rt = 0; rt < 4; ++rt) acc[rt] = (v8f)0.0f;

        for (int k = 0; k < MLEN; k += 4) {
            const int kk = k + half * 2;
            const int kp = (k >> 1) + half;
            v2f bfrag;                          // B reused across 4 row tiles
            bfrag.x = Wb1[(size_t)kk * HBR + n];
            bfrag.y = Wb1[(size_t)(kk + 1) * HBR + n];
            #pragma unroll
            for (int rt = 0; rt < 4; ++rt) {
                const v2f afrag = XsTp[kp][rt * 16 + l16];  // one b64 DS load
                acc[rt] = __builtin_amdgcn_wmma_f32_16x16x4_f32(
                    false, afrag, false, bfrag, (short)0, acc[rt], false, false);
            }
        }
        // bias + relu -> Hs (half-waves hit disjoint bank sets)
        #pragma unroll
        for (int rt = 0; rt < 4; ++rt) {
            #pragma unroll
            for (int i = 0; i < 8; ++i) {
                const int row = rt * 16 + i + half * 8;
                Hs[row][n] = fmaxf(acc[rt][i] + bias, 0.0f);
            }
        }
    }
    __syncthreads();

    // z phase: 4 threads per row, each owns 128 of the 512 columns (float4)
    {
        const int zrow = tid & 63;
        const int q    = tid >> 6;
        const float* hrow = &Hs[zrow][q * 128];
        const float* wrow = &Wb2s[q * 256];
        float z0 = 0.0f, z1 = 0.0f;
        #pragma unroll 8
        for (int jj = 0; jj < 128; jj += 4) {
            const float4 h4 = *(const float4*)(hrow + jj);
            const float4 wa = *(const float4*)(wrow + jj * 2);      // cols jj, jj+1
            const float4 wb = *(const float4*)(wrow + jj * 2 + 4);  // cols jj+2, jj+3
            z0 += h4.x * wa.x + h4.y * wa.z + h4.z * wb.x + h4.w * wb.z;
            z1 += h4.x * wa.y + h4.y * wa.w + h4.z * wb.y + h4.w * wb.w;
        }
        Zq[q][zrow][0] = z0;
        Zq[q][zrow][1] = z1;
    }
    __syncthreads();

    if (tid < ROWS) {
        const int row = tid;
        float z0 = bb2[0], z1 = bb2[1];
        #pragma unroll
        for (int q = 0; q < 4; ++q) { z0 += Zq[q][row][0]; z1 += Zq[q][row][1]; }
        const float mz = fmaxf(z0, z1);
        const float e0 = expf(z0 - mz);
        const float e1 = expf(z1 - mz);
        const float s  = e0 + e1;
        const float p0 = e0 / s;
        const float p1 = e1 / s;
        float best = -1.0f;
        int   bi   = 0;
        for (int leaf = 0; leaf < 32; ++leaf) {
            float g = ((leaf >> 4) & 1) ? p1 : p0;   // l = 0 (MSB first)
            #pragma unroll
            for (int l = 1; l < 5; ++l)
                g *= ((leaf >> (4 - l)) & 1) ? p1 : p0;
            if (g >= best) { best = g; bi = leaf; }  // >= : ties -> largest idx
        }
        router[r0 + row] = bi;
    }
}

// ---------------------------------------------------------------------------
// Kernel 2: hl = relu(x2 @ Wl1 + bl1) via WMMA, chunked over the 2048 hidden
// dim; each 128-col chunk is immediately consumed by the per-row gathered
// Wl2[:, router*48 : +48] accumulation (only 48/1536 columns computed).
// ---------------------------------------------------------------------------
__global__ __launch_bounds__(256) void latent_select_kernel(
    const float* __restrict__ batch,   // (B, 2, 128) -> row b is exactly x2[b]
    const float* __restrict__ Wl1,     // (256, 2048)
    const float* __restrict__ bl1,     // (2048)
    const float* __restrict__ Wl2,     // (2048, 1536)
    const float* __restrict__ bl2,     // (1536)
    const int* __restrict__ router,    // (B)
    float* __restrict__ out)           // (B, 48)
{
    __shared__ v2f   X2sTp[MLEN][80];     // x2 K-paired transpose (K=256)
    __shared__ float Hs[ROWS][132];       // hl chunk; 16B-aligned rows
    __shared__ int   Rc[ROWS];            // router*48 per row

    const int tid  = threadIdx.x;
    const int lane = tid & 31;
    const int w    = tid >> 5;
    const int r0   = blockIdx.x * ROWS;

    // stage x2 (= contiguous batch row: cam0 | cam1) K-paired transposed
    #pragma unroll
    for (int i = 0; i < 16; ++i) {
        const int f  = tid + i * 256;      // 4096 float4s total
        const int m  = f >> 6;             // 64 float4 per row
        const int c4 = f & 63;
        const float4 v = *(const float4*)(batch + (size_t)(r0 + m) * 256 + c4 * 4);
        X2sTp[c4 * 2 + 0][m] = (v2f){v.x, v.y};
        X2sTp[c4 * 2 + 1][m] = (v2f){v.z, v.w};
    }
    if (tid < ROWS) Rc[tid] = router[r0 + tid] * LATENT;
    __syncthreads();

    const int half   = lane >> 4;
    const int l16    = lane & 15;
    const int selRow = tid >> 2;          // 4 threads per row
    const int jb     = (tid & 3) * 12;    // 12 output cols per thread
    const int colbase = Rc[selRow];

    float acc12[12];
    #pragma unroll
    for (int j = 0; j < 12; ++j) acc12[j] = bl2[colbase + jb + j];

    for (int c = 0; c < HLAT / 128; ++c) {
        // --- WMMA: wave w owns chunk cols [w*16, w*16+16), all 4 row tiles ---
        const int gcol = c * 128 + w * 16 + l16;
        v8f acc[4];
        #pragma unroll
        for (int rt = 0; rt < 4; ++rt) acc[rt] = (v8f)0.0f;

        for (int k = 0; k < 2 * MLEN; k += 4) {
            const int kk = k + half * 2;
            const int kp = (k >> 1) + half;
            v2f bfrag;                      // B reused across 4 row tiles
            bfrag.x = Wl1[(size_t)kk * HLAT + gcol];
            bfrag.y = Wl1[(size_t)(kk + 1) * HLAT + gcol];
            #pragma unroll
            for (int rt = 0; rt < 4; ++rt) {
                const v2f afrag = X2sTp[kp][rt * 16 + l16];  // one b64 DS load
                acc[rt] = __builtin_amdgcn_wmma_f32_16x16x4_f32(
                    false, afrag, false, bfrag, (short)0, acc[rt], false, false);
            }
        }
        const float bias = bl1[gcol];

        __syncthreads();                   // previous select done reading Hs
        const int colc = w * 16 + l16;
        #pragma unroll
        for (int rt = 0; rt < 4; ++rt) {
            #pragma unroll
            for (int i = 0; i < 8; ++i) {
                const int row = rt * 16 + i + half * 8;
                Hs[row][colc] = fmaxf(acc[rt][i] + bias, 0.0f);
            }
        }
        __syncthreads();                   // Hs chunk ready

        // --- select: out[row] += Hs[row] @ Wl2[chunk, colbase+jb : +12] ---
        const float* w2 = Wl2 + (size_t)(c * 128) * OUTW + colbase + jb;
        if (c + 1 < HLAT / 128)            // warm L2 for next chunk's stream
            __builtin_prefetch(w2 + (size_t)128 * OUTW, 0, 1);

        #pragma unroll 2
        for (int k = 0; k < 128; k += 4) {
            const float4 h4 = *(const float4*)(&Hs[selRow][k]);  // 1 b128 DS ld
            #pragma unroll
            for (int kk = 0; kk < 4; ++kk) {
                const float h = (kk == 0) ? h4.x : (kk == 1) ? h4.y
                               : (kk == 2) ? h4.z : h4.w;
                const float* wr = w2 + (size_t)(k + kk) * OUTW;
                const float4 a = *(const float4*)(wr + 0);
                const float4 b = *(const float4*)(wr + 4);
                const float4 d = *(const float4*)(wr + 8);
                acc12[0]  += h * a.x;  acc12[1]  += h * a.y;
                acc12[2]  += h * a.z;  acc12[3]  += h * a.w;
                acc12[4]  += h * b.x;  acc12[5]  += h * b.y;
                acc12[6]  += h * b.z;  acc12[7]  += h * b.w;
                acc12[8]  += h * d.x;  acc12[9]  += h * d.y;
                acc12[10] += h * d.z;  acc12[11] += h * d.w;
            }
        }
    }

    float* o = out + (size_t)(r0 + selRow) * LATENT + jb;
    #pragma unroll
    for (int j = 0; j < 12; ++j) o[j] = acc12[j];
}

// ---------------------------------------------------------------------------
extern "C" void kernel_launch(void* const* d_in, const int* in_sizes, int n_in,
                              void* d_out, int out_size, void* d_ws, size_t ws_size,
                              hipStream_t stream) {
    const float* batch = (const float*)d_in[0];
    const float* Wl1   = (const float*)d_in[1];
    const float* bl1   = (const float*)d_in[2];
    const float* Wl2   = (const float*)d_in[3];
    const float* bl2   = (const float*)d_in[4];
    const float* Wb1   = (const float*)d_in[5];
    const float* bb1   = (const float*)d_in[6];
    const float* Wb2   = (const float*)d_in[7];
    const float* bb2   = (const float*)d_in[8];
    float* out   = (float*)d_out;
    int* routeri = (int*)d_ws;             // 16384 ints of scratch

    const int nblk = 16384 / ROWS;         // 256 blocks, 256 threads (8 waves)
    router_kernel<<<nblk, 256, 0, stream>>>(batch, Wb1, bb1, Wb2, bb2, routeri);
    latent_select_kernel<<<nblk, 256, 0, stream>>>(batch, Wl1, bl1, Wl2, bl2,
                                                   routeri, out);
}